// QuConvSXZ_1297080123412
// MI455X (gfx1250) — compile-verified
//
#include <hip/hip_runtime.h>
#include <hip/hip_bf16.h>

typedef __attribute__((ext_vector_type(2))) float v2f;
typedef __attribute__((ext_vector_type(8))) float v8f;

#define NQ 12
#define QDIM 4096
#define NPAIR 6
#define W_MUL 0.6324555320336759f  // sqrt(2/5)

__device__ __forceinline__ float2 cmulf(float2 a, float2 b) {
    return make_float2(a.x * b.x - a.y * b.y, a.x * b.y + a.y * b.x);
}

__global__ __launch_bounds__(256) void quconv_kernel(const float* __restrict__ x,
                                                     const float* __restrict__ weight,
                                                     float* __restrict__ out,
                                                     int writeComplex) {
    __shared__ float sRe[QDIM];
    __shared__ float sIm[QDIM];
    __shared__ float2 sG[NPAIR][16];  // 4x4 complex gate per pair, row-major

    const int tid = threadIdx.x;
    const int b = blockIdx.x;

    // ---- Phase 0: build G_p = Rzz @ Ryy @ Rxx  (one thread per pair) ----
    if (tid < NPAIR) {
        const int p = tid;
        const float th0 = weight[36 + 6 * p + 0] * W_MUL;
        const float th1 = weight[36 + 6 * p + 1] * W_MUL;
        const float th2 = weight[36 + 6 * p + 2] * W_MUL;
        const float c0 = cosf(0.5f * th0), s0 = sinf(0.5f * th0);
        const float c1 = cosf(0.5f * th1), s1 = sinf(0.5f * th1);
        const float c2 = cosf(0.5f * th2), s2 = sinf(0.5f * th2);

        const float ys[4] = {-1.f, 1.f, 1.f, -1.f};  // YY antidiag
        const float zs[4] = {1.f, -1.f, -1.f, 1.f};  // ZZ diag

        float2 Rxx[16], Ryy[16];
#pragma unroll
        for (int k = 0; k < 16; ++k) {
            Rxx[k] = make_float2(0.f, 0.f);
            Ryy[k] = make_float2(0.f, 0.f);
        }
#pragma unroll
        for (int k = 0; k < 4; ++k) {
            Rxx[k * 4 + k] = make_float2(c0, 0.f);
            Rxx[k * 4 + (3 - k)] = make_float2(0.f, -s0);
            Ryy[k * 4 + k] = make_float2(c1, 0.f);
            Ryy[k * 4 + (3 - k)] = make_float2(0.f, -s1 * ys[k]);
        }
        // T = Ryy @ Rxx
        float2 T[16];
#pragma unroll
        for (int i = 0; i < 4; ++i) {
#pragma unroll
            for (int j = 0; j < 4; ++j) {
                float2 acc = make_float2(0.f, 0.f);
#pragma unroll
                for (int k = 0; k < 4; ++k) {
                    float2 m = cmulf(Ryy[i * 4 + k], Rxx[k * 4 + j]);
                    acc.x += m.x;
                    acc.y += m.y;
                }
                T[i * 4 + j] = acc;
            }
        }
        // G = diag(Rzz) @ T ;  Rzz[i][i] = c2 - i*s2*zs[i]
#pragma unroll
        for (int i = 0; i < 4; ++i) {
            float2 dz = make_float2(c2, -s2 * zs[i]);
#pragma unroll
            for (int j = 0; j < 4; ++j) {
                sG[p][i * 4 + j] = cmulf(dz, T[i * 4 + j]);
            }
        }
    }

    // ---- Phase 1: load psi (real) into LDS ----
    for (int r = tid; r < QDIM; r += 256) {
        sRe[r] = x[(size_t)b * QDIM + r];
        sIm[r] = 0.f;
    }
    __syncthreads();

    // ---- Phase 2: phi = (G0 (x) G1 (x) ... (x) G5) psi, sweep per pair ----
    for (int p = 0; p < NPAIR; ++p) {
        const int s = 2 * (5 - p);        // bit position of this pair's base-4 digit
        const int lmask = (1 << s) - 1;

        float inr[4][4], ini[4][4];
        int bas[4];
#pragma unroll
        for (int gi = 0; gi < 4; ++gi) {
            int g = tid + (gi << 8);                       // group id, 1024 groups total
            int base = ((g & ~lmask) << 2) | (g & lmask);  // insert digit=0 at bits [s+1:s]
            bas[gi] = base;
#pragma unroll
            for (int e = 0; e < 4; ++e) {
                int idx = base + (e << s);
                inr[gi][e] = sRe[idx];
                ini[gi][e] = sIm[idx];
            }
        }
        __syncthreads();
#pragma unroll
        for (int gi = 0; gi < 4; ++gi) {
#pragma unroll
            for (int d = 0; d < 4; ++d) {
                float ar = 0.f, ai = 0.f;
#pragma unroll
                for (int e = 0; e < 4; ++e) {
                    float2 g2 = sG[p][d * 4 + e];
                    ar += g2.x * inr[gi][e] - g2.y * ini[gi][e];
                    ai += g2.x * ini[gi][e] + g2.y * inr[gi][e];
                }
                int idx = bas[gi] + (d << s);
                sRe[idx] = ar;
                sIm[idx] = ai;
            }
        }
        __syncthreads();
    }

    // ---- Phase 3: red = M^T @ conj(M), M[t][a] = phi[t*64+a]  (64x64x64, WMMA f32) ----
    // Complex: re = Mre^T Mre + Mim^T Mim ; im = Mim^T Mre - Mre^T Mim
    const int wave = tid >> 5;
    const int lane = tid & 31;
    const int mN = lane & 15;           // A-row (M) / B-col (N) for this lane
    const int kb = (lane >> 4) << 1;    // lanes 0-15 -> K 0/1 ; lanes 16-31 -> K 2/3

    for (int t = wave; t < 16; t += 8) {  // 16 tiles, 2 per wave, uniform per wave
        const int i0 = (t >> 2) << 4;
        const int j0 = (t & 3) << 4;
        v8f cRR = {0.f, 0.f, 0.f, 0.f, 0.f, 0.f, 0.f, 0.f};
        v8f cII = cRR, cIR = cRR, cRI = cRR;

        for (int kk = 0; kk < 64; kk += 4) {
            const int r0 = (kk + kb) << 6;  // row (kk+kb) of M, row-major stride 64
            const int r1 = r0 + 64;
            // A-frag = X^T[i0..i0+15][kk..kk+3]  => per-lane X[kk+kb+{0,1}][i0+m]
            v2f aR = {sRe[r0 + i0 + mN], sRe[r1 + i0 + mN]};
            v2f aI = {sIm[r0 + i0 + mN], sIm[r1 + i0 + mN]};
            // B-frag = Y[kk..kk+3][j0..j0+15]    => per-lane Y[kk+kb+{0,1}][j0+n]
            v2f bR = {sRe[r0 + j0 + mN], sRe[r1 + j0 + mN]};
            v2f bI = {sIm[r0 + j0 + mN], sIm[r1 + j0 + mN]};

            cRR = __builtin_amdgcn_wmma_f32_16x16x4_f32(false, aR, false, bR, (short)0, cRR, false, false);
            cII = __builtin_amdgcn_wmma_f32_16x16x4_f32(false, aI, false, bI, (short)0, cII, false, false);
            cIR = __builtin_amdgcn_wmma_f32_16x16x4_f32(false, aI, false, bR, (short)0, cIR, false, false);
            cRI = __builtin_amdgcn_wmma_f32_16x16x4_f32(false, aR, false, bI, (short)0, cRI, false, false);
        }

        // C/D layout: VGPR v -> M = v + 8*(lane>=16), N = lane&15
        const int hi = lane >> 4;
#pragma unroll
        for (int v = 0; v < 8; ++v) {
            int a1 = i0 + v + (hi << 3);
            int a2 = j0 + mN;
            float re = cRR[v] + cII[v];
            float im = cIR[v] - cRI[v];
            size_t o = (size_t)b * QDIM + (size_t)(a1 << 6) + a2;
            if (writeComplex) {
                out[2 * o + 0] = re;
                out[2 * o + 1] = im;
            } else {
                out[o] = re;
            }
        }
    }
}

extern "C" void kernel_launch(void* const* d_in, const int* in_sizes, int n_in,
                              void* d_out, int out_size, void* d_ws, size_t ws_size,
                              hipStream_t stream) {
    const float* x = (const float*)d_in[0];       // [B,1,4096] f32
    const float* w = (const float*)d_in[1];       // [72] f32
    (void)d_ws; (void)ws_size; (void)n_in;

    const int B = in_sizes[0] / QDIM;
    const int writeComplex = (out_size >= B * QDIM * 2) ? 1 : 0;

    quconv_kernel<<<dim3(B), dim3(256), 0, stream>>>(x, w, (float*)d_out, writeComplex);
}